// PersistenceLoss_36429912605420
// MI455X (gfx1250) — compile-verified
//
#include <hip/hip_runtime.h>
#include <hip/hip_bf16.h>
#include <cstdint>

// ---------------------------------------------------------------------------
// PersistenceLoss: bce(y_true,y_pred) + 0.005 * topo
//   topo = mean over 4096 patches of mean((sort(pred_patch)-sort(true_patch))^2)
//   Both terms share the denominator 2^24:
//       out = (sum_bce_terms + 0.005 * sum_sq_sorted_diffs) * 2^-24
//
// One workgroup per 64x64 patch. The strided 2D patch (64 rows x 256B,
// row stride 2048B) is pulled into LDS by the CDNA5 Tensor Data Mover
// (tensor_load_to_lds, TENSORcnt). Hybrid bitonic sort: stages with j<=8
// run in registers on thread-private 16-element chunks; only j>=16 stages
// touch LDS (36 LDS passes instead of 78, ~45 barriers instead of 78).
// ---------------------------------------------------------------------------

typedef unsigned int u32x4 __attribute__((ext_vector_type(4)));
typedef int          i32x8 __attribute__((ext_vector_type(8)));
typedef int          i32x4 __attribute__((ext_vector_type(4)));

#define IMG_W   512
#define TS      64          // patch side
#define PATCH_N 4096        // TS*TS elements per patch
#define NBLOCKS 4096        // 64 batch * 64 patches
#define NTHREADS 256
#define INV_TOTAL 5.9604644775390625e-8f   // 1 / 2^24

// Issue a TDM 2D tile load: 64x64 fp32 tile, row stride 512 elements,
// destination = LDS byte offset lds_off. Wave-level op (EXEC ignored).
__device__ __forceinline__ void tdm_load_patch(uint32_t lds_off, uint64_t gaddr)
{
    // ---- D# group 0 (128b): count=1 | lds_addr | global_addr[56:0] | type=2
    u32x4 g0;
    g0.x = 1u;                                   // count=1, user descriptor
    g0.y = lds_off;                              // LDS byte address
    g0.z = (uint32_t)(gaddr & 0xFFFFFFFFull);    // global_addr[31:0]
    g0.w = (uint32_t)((gaddr >> 32) & 0x01FFFFFFull) | (2u << 30); // [56:32]|type=2

    // ---- D# group 1 (256b)
    const uint64_t tdim0   = 512;                // tensor width  (elements)
    const uint64_t tdim1   = 1u << 20;           // tensor height (generous)
    const uint64_t tile0   = TS;                 // tile width
    const uint64_t tile1   = TS;                 // tile height
    const uint64_t stride0 = IMG_W;              // row stride (elements)
    union { unsigned long long q[4]; i32x8 v; } g1;
    g1.q[0] = (2ull << 16)                       // data_size = 4 bytes
            | ((tdim0 & 0xFFFFull) << 48);       // tensor_dim0[15:0]
    g1.q[1] = ((tdim0 >> 16) & 0xFFFFull)        // tensor_dim0[31:16]
            | (tdim1 << 16)                      // tensor_dim1
            | (tile0 << 48);                     // tile_dim0
    g1.q[2] = tile1                              // tile_dim1
            | ((stride0 & 0xFFFFFFFFull) << 32); // tensor_dim0_stride[31:0]
    g1.q[3] = ((stride0 >> 32) & 0xFFFFull);     // stride0[47:32]; stride1=0

    const i32x4 zero4 = {0, 0, 0, 0};            // groups 2/3 unused (2D tile)
    const i32x8 zero8 = {0, 0, 0, 0, 0, 0, 0, 0};
    __builtin_amdgcn_tensor_load_to_lds(g0, g1.v, zero4, zero4, zero8, 0);
}

__device__ __forceinline__ void cmpswap(float& a, float& b, bool up)
{
    const float lo = fminf(a, b);
    const float hi = fmaxf(a, b);
    a = up ? lo : hi;
    b = up ? hi : lo;
}

// Full bitonic network k=2..16 on a register chunk of 16; base = global index
// of r[0] (multiple of 16), used only for the k=16 direction bit.
__device__ __forceinline__ void sort16(float r[16], unsigned base)
{
#pragma unroll
    for (unsigned k = 2; k <= 16; k <<= 1) {
#pragma unroll
        for (unsigned j = k >> 1; j > 0; j >>= 1) {
#pragma unroll
            for (unsigned li = 0; li < 16; ++li) {
                if ((li & j) == 0u) {
                    const bool up = (((base | li) & k) == 0u);  // no carry: base%16==0
                    cmpswap(r[li], r[li | j], up);
                }
            }
        }
    }
}

// Bitonic merge stages j=8,4,2,1, uniform direction (valid when k>=32).
__device__ __forceinline__ void merge16(float r[16], bool up)
{
#pragma unroll
    for (unsigned j = 8; j > 0; j >>= 1) {
#pragma unroll
        for (unsigned li = 0; li < 16; ++li)
            if ((li & j) == 0u) cmpswap(r[li], r[li | j], up);
    }
}

__device__ __forceinline__ void load16(const float* lds, float r[16])
{
    const float4* p = (const float4*)lds;        // 64B-aligned (base*4)
#pragma unroll
    for (int v = 0; v < 4; ++v) {
        const float4 q = p[v];
        r[4 * v + 0] = q.x; r[4 * v + 1] = q.y;
        r[4 * v + 2] = q.z; r[4 * v + 3] = q.w;
    }
}

__device__ __forceinline__ void store16(float* lds, const float r[16])
{
    float4* p = (float4*)lds;
#pragma unroll
    for (int v = 0; v < 4; ++v)
        p[v] = make_float4(r[4 * v + 0], r[4 * v + 1], r[4 * v + 2], r[4 * v + 3]);
}

__global__ __launch_bounds__(NTHREADS)
void persistence_patch_kernel(const float* __restrict__ y_true,
                              const float* __restrict__ y_pred,
                              float* __restrict__ partials)
{
    __shared__ float tp[PATCH_N];   // true patch (gets sorted)
    __shared__ float pp[PATCH_N];   // pred patch (gets sorted)
    __shared__ float red[NTHREADS];

    const unsigned tid  = threadIdx.x;
    const unsigned blk  = blockIdx.x;
    const unsigned base = tid * 16u;             // thread-owned chunk [base, base+16)

    // patch coordinates: blk = b*64 + py*8 + px
    const unsigned b  = blk >> 6;
    const unsigned pi = blk & 63u;
    const unsigned py = pi >> 3, px = pi & 7u;
    const size_t elem0 = (size_t)b * (IMG_W * IMG_W)
                       + (size_t)(py * TS) * IMG_W
                       + (size_t)(px * TS);

    // ---- async tile load via Tensor Data Mover (wave 0 only) ----
    if (tid < 32u) {
        tdm_load_patch((uint32_t)(uintptr_t)(void*)tp,
                       (uint64_t)(uintptr_t)(y_true + elem0));
        tdm_load_patch((uint32_t)(uintptr_t)(void*)pp,
                       (uint64_t)(uintptr_t)(y_pred + elem0));
        __builtin_amdgcn_s_wait_tensorcnt(0);
    }
    __syncthreads();

    float r[16], s[16];
    load16(&tp[base], r);
    load16(&pp[base], s);

    // ---- BCE over the chunk (positionally paired -> before any sorting) ----
    float bce = 0.0f;
#pragma unroll
    for (int li = 0; li < 16; ++li) {
        const float y = r[li];
        const float p = fminf(fmaxf(s[li], 1e-7f), 1.0f - 1e-7f);
        bce -= y * logf(p) + (1.0f - y) * log1pf(-p);
    }

    // ---- phase 1: k = 2..16 entirely in registers ----
    sort16(r, base);
    sort16(s, base);
    store16(&tp[base], r);
    store16(&pp[base], s);
    __syncthreads();

    // ---- phase 2: k = 32..4096 ----
    for (unsigned k = 32; k <= PATCH_N; k <<= 1) {
        // cross-chunk stages through LDS
        for (unsigned j = k >> 1; j >= 16; j >>= 1) {
            for (unsigned t = tid; t < (PATCH_N / 2); t += NTHREADS) {
                const unsigned i  = ((t & ~(j - 1u)) << 1) | (t & (j - 1u));
                const unsigned ix = i | j;
                const bool up = ((i & k) == 0u);
                cmpswap(tp[i], tp[ix], up);
                cmpswap(pp[i], pp[ix], up);
            }
            __syncthreads();
        }
        // tail stages j=8..1 fused in registers (direction uniform per chunk)
        const bool up = ((base & k) == 0u);
        load16(&tp[base], r);
        load16(&pp[base], s);
        merge16(r, up);
        merge16(s, up);
        if (k < PATCH_N) {                       // final chunk stays in registers
            store16(&tp[base], r);
            store16(&pp[base], s);
            __syncthreads();
        }
    }

    // ---- squared distance between aligned sorted values (from registers) ----
    float sq = 0.0f;
#pragma unroll
    for (int li = 0; li < 16; ++li) {
        const float d = s[li] - r[li];
        sq += d * d;
    }

    // ---- block reduction of fused partial ----
    red[tid] = bce + 0.005f * sq;
    __syncthreads();
    for (unsigned off = NTHREADS / 2; off > 0; off >>= 1) {
        if (tid < off) red[tid] += red[tid + off];
        __syncthreads();
    }
    if (tid == 0) partials[blk] = red[0];
}

__global__ __launch_bounds__(NTHREADS)
void persistence_final_kernel(const float* __restrict__ partials,
                              float* __restrict__ out)
{
    __shared__ float red[NTHREADS];
    const unsigned tid = threadIdx.x;
    float sum = 0.0f;
    for (unsigned i = tid; i < NBLOCKS; i += NTHREADS) sum += partials[i];
    red[tid] = sum;
    __syncthreads();
    for (unsigned off = NTHREADS / 2; off > 0; off >>= 1) {
        if (tid < off) red[tid] += red[tid + off];
        __syncthreads();
    }
    if (tid == 0) out[0] = red[0] * INV_TOTAL;
}

extern "C" void kernel_launch(void* const* d_in, const int* in_sizes, int n_in,
                              void* d_out, int out_size, void* d_ws, size_t ws_size,
                              hipStream_t stream)
{
    const float* y_true = (const float*)d_in[0];
    const float* y_pred = (const float*)d_in[1];
    float* partials = (float*)d_ws;          // 4096 floats = 16 KB scratch
    float* out = (float*)d_out;

    persistence_patch_kernel<<<NBLOCKS, NTHREADS, 0, stream>>>(y_true, y_pred, partials);
    persistence_final_kernel<<<1, NTHREADS, 0, stream>>>(partials, out);
}